// Global_Att_37855841747265
// MI455X (gfx1250) — compile-verified
//
#include <hip/hip_runtime.h>
#include <hip/hip_bf16.h>
#include <stdint.h>

// ---------------------------------------------------------------------------
// Global point-cloud attention for MI455X (gfx1250), f32 WMMA path.
//
// B=2 N=2048 C=256 H=8 HD=32.  Distance bias folded into the QK^T GEMM by
// augmenting the K-dimension to 40:  Q' = [q*scale, 2x,2y,2z, -|x|^2, -1, 0,0,0]
//                                    K' = [k,       x, y, z,   1, |x|^2, 0,0,0]
// so Q'.K' = q.k*scale - dist.  Flash-attention over 16-key chunks with
// double-buffered ASYNC global->LDS DMA; logits tensor (268 MB) is never
// materialized.  Everything stays L2-resident (192 MB).
// ---------------------------------------------------------------------------

#define BB   2
#define NN   2048
#define CC   256
#define HH   8
#define HD   32
#define KA   40                       // augmented K dimension (32 + 5, pad to 40)
#define SCALE 0.17677669529663687f    // 1/sqrt(32)

typedef __attribute__((ext_vector_type(2))) float v2f;
typedef __attribute__((ext_vector_type(8))) float v8f;

// D(16x16,f32) = A(16x4,f32) * B(4x16,f32) + C   — V_WMMA_F32_16X16X4_F32
__device__ __forceinline__ v8f wmma4(v2f a, v2f b, v8f c) {
  return __builtin_amdgcn_wmma_f32_16x16x4_f32(
      /*neg_a=*/false, a, /*neg_b=*/false, b,
      /*c_mod=*/(short)0, c, /*reuse_a=*/false, /*reuse_b=*/false);
}

// Async DMA: global -> LDS, 16 bytes, tracked by ASYNCcnt (CDNA5).
// Generic shared pointer truncated to 32 bits == LDS byte offset (ISA 10.2).
__device__ __forceinline__ void g2lds_b128(const float* lds_dst, const float* gsrc) {
  unsigned loff = (unsigned)(uintptr_t)lds_dst;
  unsigned long long ga = (unsigned long long)(uintptr_t)gsrc;
  asm volatile("global_load_async_to_lds_b128 %0, %1, off"
               :: "v"(loff), "v"(ga) : "memory");
}
__device__ __forceinline__ void wait_async0() {
  asm volatile("s_wait_asynccnt 0x0" ::: "memory");
}

// ---------------------------------------------------------------------------
// Kernel 1: fused QKV projection.  grid = (4096/64, 3), block = 256 (8 waves).
// WG tile 64 rows x 256 cols; wave tile 32x64 (2 M-subtiles reuse B frags).
// Register-pipelined: iteration i+1 fragments load during iteration i WMMAs.
// ---------------------------------------------------------------------------
__device__ __forceinline__ void ld_frags_gemm(
    const float* a0, const float* a1, const float* __restrict__ W,
    int colBase, int lmod, int ki, v2f af[2], v2f bf[4]) {
  af[0] = *(const v2f*)(a0 + ki);
  af[1] = *(const v2f*)(a1 + ki);
#pragma unroll
  for (int t = 0; t < 4; ++t) {
    const int col = colBase + t * 16 + lmod;
    bf[t].x = W[(long)ki * CC + col];
    bf[t].y = W[(long)(ki + 1) * CC + col];
  }
}

__global__ void __launch_bounds__(256) ga_qkv_kernel(
    const float* __restrict__ feat,
    const float* __restrict__ Wq, const float* __restrict__ bq,
    const float* __restrict__ Wk, const float* __restrict__ bk,
    const float* __restrict__ Wv, const float* __restrict__ bv,
    float* __restrict__ qa, float* __restrict__ kaug, float* __restrict__ vv) {
  const int which = blockIdx.y;                     // 0=q, 1=k, 2=v
  const float* __restrict__ W    = (which == 0) ? Wq : (which == 1) ? Wk : Wv;
  const float* __restrict__ bias = (which == 0) ? bq : (which == 1) ? bk : bv;

  const int tid   = threadIdx.x;
  const int wave  = tid >> 5;
  const int lane  = tid & 31;
  const int lmod  = lane & 15;
  const int lhalf = lane >> 4;
  const int koff  = lhalf * 2;
  const int rowBase = blockIdx.x * 64 + (wave >> 2) * 32;   // wave: 32 rows
  const int colBase = (wave & 3) * 64;                      // wave: 64 cols

  const float* a0 = feat + (long)(rowBase + lmod) * CC;
  const float* a1 = a0 + 16 * CC;

  v8f acc[2][4];
#pragma unroll
  for (int g = 0; g < 2; ++g)
#pragma unroll
    for (int t = 0; t < 4; ++t) acc[g][t] = v8f{};

  v2f afc[2], bfc[4], afn[2], bfn[4];
  ld_frags_gemm(a0, a1, W, colBase, lmod, koff, afc, bfc);

  for (int kk = 0; kk < CC; kk += 4) {
    const int kin = ((kk + 4) & (CC - 1)) + koff;   // wraps on last iter (discarded)
    ld_frags_gemm(a0, a1, W, colBase, lmod, kin, afn, bfn);
#pragma unroll
    for (int t = 0; t < 4; ++t) acc[0][t] = wmma4(afc[0], bfc[t], acc[0][t]);
#pragma unroll
    for (int t = 0; t < 4; ++t) acc[1][t] = wmma4(afc[1], bfc[t], acc[1][t]);
#pragma unroll
    for (int g = 0; g < 2; ++g) afc[g] = afn[g];
#pragma unroll
    for (int t = 0; t < 4; ++t) bfc[t] = bfn[t];
  }

  // Epilogue: C/D layout -> row m = base + r + 8*lhalf, col = colBase+t*16+lmod
#pragma unroll
  for (int g = 0; g < 2; ++g) {
#pragma unroll
    for (int r = 0; r < 8; ++r) {
      const int m = rowBase + g * 16 + r + 8 * lhalf;
      const int b = m / NN;
      const int n = m % NN;
#pragma unroll
      for (int t = 0; t < 4; ++t) {
        const int col = colBase + t * 16 + lmod;
        const float val = acc[g][t][r] + bias[col];
        const int h = col >> 5, d = col & 31;
        const long hm = ((long)(b * HH + h) * NN + n);
        if (which == 0)      qa  [hm * KA + d] = val * SCALE;
        else if (which == 1) kaug[hm * KA + d] = val;
        else                 vv  [hm * HD + d] = val;
      }
    }
  }
}

// ---------------------------------------------------------------------------
// Kernel 2: geometric tail columns of the augmented Q/K.
// ---------------------------------------------------------------------------
__global__ void __launch_bounds__(256) ga_augtail_kernel(
    const float* __restrict__ pts, float* __restrict__ qa, float* __restrict__ kaug) {
  const int idx = blockIdx.x * blockDim.x + threadIdx.x;    // (b*H+h)*N + n
  if (idx >= BB * HH * NN) return;
  const int n = idx % NN;
  const int b = (idx / NN) / HH;
  const float x = pts[(long)(b * NN + n) * 3 + 0];
  const float y = pts[(long)(b * NN + n) * 3 + 1];
  const float z = pts[(long)(b * NN + n) * 3 + 2];
  const float s2 = x * x + y * y + z * z;
  float* qt = qa + (long)idx * KA + HD;
  qt[0] = 2.f * x; qt[1] = 2.f * y; qt[2] = 2.f * z;
  qt[3] = -s2;     qt[4] = -1.f;    qt[5] = 0.f; qt[6] = 0.f; qt[7] = 0.f;
  float* kt = kaug + (long)idx * KA + HD;
  kt[0] = x; kt[1] = y; kt[2] = z;
  kt[3] = 1.f; kt[4] = s2; kt[5] = 0.f; kt[6] = 0.f; kt[7] = 0.f;
}

// ---------------------------------------------------------------------------
// Kernel 3: flash attention.  grid = (N/64, B*H), block = 128 (4 waves).
// Double-buffered ASYNC global->LDS staging of 16-key K'/V chunks overlapped
// with the 18-WMMA compute; one barrier per chunk.  P staged through
// wave-private LDS to convert C/D accumulator layout -> A-frag layout.
// ---------------------------------------------------------------------------
__global__ void __launch_bounds__(128) ga_attn_kernel(
    const float* __restrict__ qa, const float* __restrict__ kaug,
    const float* __restrict__ vv, float* __restrict__ ao) {
  __shared__ __align__(16) float Ks[2][16][KA];   // 5.0 KB
  __shared__ __align__(16) float Vs[2][16][HD];   // 4.0 KB
  __shared__ __align__(16) float Ps[4][16][16];   // 4.0 KB (per-wave P staging)

  const int tid   = threadIdx.x;
  const int wave  = tid >> 5;
  const int lane  = tid & 31;
  const int lmod  = lane & 15;
  const int lhalf = lane >> 4;
  const int koff  = lhalf * 2;
  const int bh    = blockIdx.y;
  const int qBase = blockIdx.x * 64 + wave * 16;

  const float* __restrict__ kb_ptr = kaug + (long)bh * NN * KA;
  const float* __restrict__ vb_ptr = vv   + (long)bh * NN * HD;
  const float* __restrict__ qrow   = qa + ((long)bh * NN + qBase + lmod) * KA;

  v2f aq[10];
#pragma unroll
  for (int j = 0; j < 10; ++j) aq[j] = *(const v2f*)(qrow + j * 4 + koff);

  v8f o0 = {}, o1 = {};
  float mrun[8], lrun[8];
#pragma unroll
  for (int r = 0; r < 8; ++r) { mrun[r] = -3.0e38f; lrun[r] = 0.f; }

  // Async-stage one 16-key chunk (K': 640 f32, V: 512 f32, both contiguous).
  auto stage = [&](int buf, int kb) {
    const float* gk = kb_ptr + (long)kb * KA;
    const float* gv = vb_ptr + (long)kb * HD;
    for (int i = tid; i < (16 * KA) / 4; i += 128)
      g2lds_b128(&Ks[buf][0][0] + i * 4, gk + i * 4);
    for (int i = tid; i < (16 * HD) / 4; i += 128)
      g2lds_b128(&Vs[buf][0][0] + i * 4, gv + i * 4);
  };

  stage(0, 0);
  wait_async0();
  __syncthreads();

  int cur = 0;
  for (int kb = 0; kb < NN; kb += 16, cur ^= 1) {
    if (kb + 16 < NN) stage(cur ^ 1, kb + 16);    // DMA next chunk during compute

    // S(16x16) = Q'(16x40) x K'^T(40x16): 10 WMMAs (frags batched first).
    v2f bfs[10];
#pragma unroll
    for (int j = 0; j < 10; ++j)
      bfs[j] = *(const v2f*)(&Ks[cur][lmod][j * 4 + koff]);
    v8f s = {};
#pragma unroll
    for (int j = 0; j < 10; ++j) s = wmma4(aq[j], bfs[j], s);

    // Online softmax; each row lives across a 16-lane half of the wave.
#pragma unroll
    for (int r = 0; r < 8; ++r) {
      float sv = s[r];
      float mx = sv;
      mx = fmaxf(mx, __shfl_xor(mx, 1));
      mx = fmaxf(mx, __shfl_xor(mx, 2));
      mx = fmaxf(mx, __shfl_xor(mx, 4));
      mx = fmaxf(mx, __shfl_xor(mx, 8));
      const float mnew = fmaxf(mrun[r], mx);
      const float p = __expf(sv - mnew);
      float ps = p;
      ps += __shfl_xor(ps, 1); ps += __shfl_xor(ps, 2);
      ps += __shfl_xor(ps, 4); ps += __shfl_xor(ps, 8);
      const float al = __expf(mrun[r] - mnew);
      lrun[r] = lrun[r] * al + ps;
      mrun[r] = mnew;
      o0[r] *= al;
      o1[r] *= al;
      Ps[wave][r + 8 * lhalf][lmod] = p;          // stage P for A-frag re-layout
    }

    // O(16x32) += P(16x16) x V(16x32): 8 WMMAs (frags batched first).
    v2f ap[4], bv0[4], bv1[4];
#pragma unroll
    for (int jj = 0; jj < 4; ++jj) {
      ap[jj] = *(const v2f*)(&Ps[wave][lmod][jj * 4 + koff]);
      const int kr = jj * 4 + koff;
      bv0[jj].x = Vs[cur][kr][lmod];      bv0[jj].y = Vs[cur][kr + 1][lmod];
      bv1[jj].x = Vs[cur][kr][16 + lmod]; bv1[jj].y = Vs[cur][kr + 1][16 + lmod];
    }
#pragma unroll
    for (int jj = 0; jj < 4; ++jj) {
      o0 = wmma4(ap[jj], bv0[jj], o0);
      o1 = wmma4(ap[jj], bv1[jj], o1);
    }

    wait_async0();        // own wave's DMA into next buffer complete
    __syncthreads();      // everyone done with cur + staged next
  }

  // Epilogue: normalize and scatter to [b][n][h*HD + d].
  const int b = bh / HH, h = bh % HH;
  float* __restrict__ aob = ao + (long)b * NN * CC;
#pragma unroll
  for (int r = 0; r < 8; ++r) {
    const float inv = 1.f / lrun[r];
    const int m = qBase + r + 8 * lhalf;
    aob[(long)m * CC + h * HD + lmod]      = o0[r] * inv;
    aob[(long)m * CC + h * HD + 16 + lmod] = o1[r] * inv;
  }
}

// ---------------------------------------------------------------------------
// Kernel 4: output projection + residual + LayerNorm.
// grid = 4096/64, block = 256 (8 waves).  WG tile 64 rows x 256 cols,
// wave tile 32x64, register-pipelined like kernel 1.
// ---------------------------------------------------------------------------
__global__ void __launch_bounds__(256) ga_outln_kernel(
    const float* __restrict__ ao, const float* __restrict__ Wo,
    const float* __restrict__ bo, const float* __restrict__ feat,
    const float* __restrict__ g, const float* __restrict__ beta,
    float* __restrict__ out) {
  __shared__ float fs[64][CC];        // 64 KB of the 320 KB WGP LDS
  __shared__ float mu_s[64], rs_s[64];

  const int tid   = threadIdx.x;
  const int wave  = tid >> 5;
  const int lane  = tid & 31;
  const int lmod  = lane & 15;
  const int lhalf = lane >> 4;
  const int koff  = lhalf * 2;
  const int mBase = (wave >> 2) * 32;           // wave: 32 rows within WG tile
  const int colBase = (wave & 3) * 64;
  const int rowBase = blockIdx.x * 64;

  const float* a0 = ao + (long)(rowBase + mBase + lmod) * CC;
  const float* a1 = a0 + 16 * CC;

  v8f acc[2][4];
#pragma unroll
  for (int gi = 0; gi < 2; ++gi)
#pragma unroll
    for (int t = 0; t < 4; ++t) acc[gi][t] = v8f{};

  v2f afc[2], bfc[4], afn[2], bfn[4];
  ld_frags_gemm(a0, a1, Wo, colBase, lmod, koff, afc, bfc);

  for (int kk = 0; kk < CC; kk += 4) {
    const int kin = ((kk + 4) & (CC - 1)) + koff;
    ld_frags_gemm(a0, a1, Wo, colBase, lmod, kin, afn, bfn);
#pragma unroll
    for (int t = 0; t < 4; ++t) acc[0][t] = wmma4(afc[0], bfc[t], acc[0][t]);
#pragma unroll
    for (int t = 0; t < 4; ++t) acc[1][t] = wmma4(afc[1], bfc[t], acc[1][t]);
#pragma unroll
    for (int gi = 0; gi < 2; ++gi) afc[gi] = afn[gi];
#pragma unroll
    for (int t = 0; t < 4; ++t) bfc[t] = bfn[t];
  }

  // residual add, stage full rows in LDS for LayerNorm
#pragma unroll
  for (int gi = 0; gi < 2; ++gi) {
#pragma unroll
    for (int r = 0; r < 8; ++r) {
      const int ml  = mBase + gi * 16 + r + 8 * lhalf;
      const int row = rowBase + ml;
#pragma unroll
      for (int t = 0; t < 4; ++t) {
        const int col = colBase + t * 16 + lmod;
        fs[ml][col] = acc[gi][t][r] + bo[col] + feat[(long)row * CC + col];
      }
    }
  }
  __syncthreads();

  if (tid < 64) {
    float s = 0.f, s2 = 0.f;
    for (int c = 0; c < CC; ++c) { float x = fs[tid][c]; s += x; s2 += x * x; }
    const float mu = s * (1.f / CC);
    const float var = s2 * (1.f / CC) - mu * mu;
    mu_s[tid] = mu;
    rs_s[tid] = rsqrtf(var + 1e-5f);
  }
  __syncthreads();

  for (int i = tid; i < 64 * CC; i += 256) {
    const int rr = i >> 8, cc = i & (CC - 1);
    out[(long)(rowBase + rr) * CC + cc] =
        (fs[rr][cc] - mu_s[rr]) * rs_s[rr] * g[cc] + beta[cc];
  }
}

// ---------------------------------------------------------------------------
extern "C" void kernel_launch(void* const* d_in, const int* in_sizes, int n_in,
                              void* d_out, int out_size, void* d_ws, size_t ws_size,
                              hipStream_t stream) {
  const float* feat = (const float*)d_in[0];
  const float* pts  = (const float*)d_in[1];
  const float* Wq   = (const float*)d_in[2];
  const float* bq   = (const float*)d_in[3];
  const float* Wk   = (const float*)d_in[4];
  const float* bk   = (const float*)d_in[5];
  const float* Wv   = (const float*)d_in[6];
  const float* bv   = (const float*)d_in[7];
  const float* Wo   = (const float*)d_in[8];
  const float* bo   = (const float*)d_in[9];
  const float* lng  = (const float*)d_in[10];
  const float* lnb  = (const float*)d_in[11];
  float* out = (float*)d_out;

  // Workspace: qa | kaug | v | attn_out  (~18 MB total, L2-resident)
  float* qa   = (float*)d_ws;
  float* kaug = qa   + (long)BB * HH * NN * KA;
  float* vv   = kaug + (long)BB * HH * NN * KA;
  float* ao   = vv   + (long)BB * HH * NN * HD;

  ga_qkv_kernel<<<dim3((BB * NN) / 64, 3), 256, 0, stream>>>(
      feat, Wq, bq, Wk, bk, Wv, bv, qa, kaug, vv);
  ga_augtail_kernel<<<(BB * HH * NN) / 256, 256, 0, stream>>>(pts, qa, kaug);
  ga_attn_kernel<<<dim3(NN / 64, BB * HH), 128, 0, stream>>>(qa, kaug, vv, ao);
  ga_outln_kernel<<<(BB * NN) / 64, 256, 0, stream>>>(
      ao, Wo, bo, feat, lng, lnb, out);
}